// VITS_Encoder_33827162423672
// MI455X (gfx1250) — compile-verified
//
#include <hip/hip_runtime.h>
#include <math.h>
#include <stdint.h>

#define B_   16
#define C_   192
#define TX_  1024
#define TY_  4096

#define TBLK 256                    // threads per block (8 waves, wave32)
#define YV   4                      // consecutive y's per thread (float4 path)
#define YTILE (TBLK * YV)           // 1024 y's per block
#define TILES_PER_B (TY_ / YTILE)   // 4
#define CSPLIT 16                   // channel groups per (b, y-tile)
#define CPER  (C_ / CSPLIT)         // 12 channels per block

#define __AS1 __attribute__((address_space(1)))
#define __AS3 __attribute__((address_space(3)))

typedef int v4i_ __attribute__((ext_vector_type(4)));

// Async-to-LDS path: only valid in the device pass, where the builtin exists
// (proven by rounds 1-2: the builtin type-checked and the call compiled).
#if defined(__HIP_DEVICE_COMPILE__) && __has_builtin(__builtin_amdgcn_global_load_async_to_lds_b128)
#define USE_ASYNC_LDS 1
#else
#define USE_ASYNC_LDS 0
#endif

// ---------------------------------------------------------------------------
// Kernel A: per-batch duration computation + inclusive scan of w_ceil.
//   cum_ws[b][x] = cumsum_x ceil(exp(logw[b,x]) * x_mask)
//   ylen_ws[b]   = clip((int)cum_ws[b][TX-1], 1, max_y)
// ---------------------------------------------------------------------------
__global__ __launch_bounds__(TX_) void vits_scan_kernel(
    const float* __restrict__ logw,      // [B,1,TX]
    const int*   __restrict__ x_lengths, // [B]
    const int*   __restrict__ max_y_p,   // [1]
    float*       __restrict__ cum_ws,    // [B,TX]
    int*         __restrict__ ylen_ws)   // [B]
{
    __shared__ float s[TX_];
    const int b = blockIdx.x;
    const int t = threadIdx.x;

    int xlen = x_lengths[b];
    if (xlen < 1) xlen = 1;

    float w = 0.0f;
    if (t < xlen) w = ceilf(expf(logw[b * TX_ + t]));   // ceil(exp(logw))*mask
    s[t] = w;
    __syncthreads();

    for (int off = 1; off < TX_; off <<= 1) {
        float v = (t >= off) ? s[t - off] : 0.0f;
        __syncthreads();
        if (t >= off) s[t] += v;
        __syncthreads();
    }

    cum_ws[b * TX_ + t] = s[t];

    if (t == 0) {
        int total = (int)s[TX_ - 1];   // exact: integer-valued float < 2^24
        int my = max_y_p[0];
        int yl = total;
        if (yl < 1)  yl = 1;
        if (yl > my) yl = my;
        ylen_ws[b] = yl;
    }
}

// ---------------------------------------------------------------------------
// Kernel B: frame expansion (one-hot gather) + noise scale + y_mask.
// grid = (B*TILES_PER_B, CSPLIT). Each thread owns YV=4 consecutive y's:
// async-stage cum into LDS, one binary search + monotonic advance, then
// stream CPER channels with b128-vectorized noise reads / z writes.
// ---------------------------------------------------------------------------
__global__ __launch_bounds__(TBLK) void vits_expand_kernel(
    const float* __restrict__ m_p,     // [B,C,TX]
    const float* __restrict__ logs_p,  // [B,C,TX]
    const float* __restrict__ noise,   // [B,C,TY]
    const float* __restrict__ cum_ws,  // [B,TX]
    const int*   __restrict__ ylen_ws, // [B]
    float*       __restrict__ z_out,   // [B,C,TY]
    float*       __restrict__ mask_out)// [B,TY]
{
    __shared__ float scum[TX_];

    const int b      = blockIdx.x / TILES_PER_B;
    const int tile   = blockIdx.x % TILES_PER_B;
    const int cgroup = blockIdx.y;
    const int y0     = tile * YTILE + threadIdx.x * YV;

    // --- async-stage cum[b][0..1023] (4 KB) into LDS ---------------------
    const float* gcum = cum_ws + (size_t)b * TX_;
    const int base = threadIdx.x * 4;       // 256 threads x 16 B = 4 KB
#if USE_ASYNC_LDS
    __builtin_amdgcn_global_load_async_to_lds_b128(
        (__AS1 v4i_*)(uintptr_t)(gcum + base),
        (__AS3 v4i_*)(uintptr_t)(&scum[base]),
        /*offset=*/0, /*cpol=*/0);
#if __has_builtin(__builtin_amdgcn_s_wait_asynccnt)
    __builtin_amdgcn_s_wait_asynccnt(0);
#else
    asm volatile("s_wait_asynccnt 0" ::: "memory");
#endif
#else
    scum[base + 0] = gcum[base + 0];
    scum[base + 1] = gcum[base + 1];
    scum[base + 2] = gcum[base + 2];
    scum[base + 3] = gcum[base + 3];
#endif
    __syncthreads();

    // --- alignment: first x with cum[x] > y, for y0..y0+3 ---------------
    const float fy0 = (float)y0;
    int lo = 0, hi = TX_;
    while (lo < hi) {
        int mid = (lo + hi) >> 1;
        if (scum[mid] > fy0) hi = mid; else lo = mid + 1;
    }

    int  xs[YV];
    bool vs[YV];
    {
        int x = lo;
#pragma unroll
        for (int j = 0; j < YV; ++j) {
            const float fy = (float)(y0 + j);
            while (x < TX_ && scum[x] <= fy) ++x;   // monotonic advance
            vs[j] = (x < TX_);
            xs[j] = vs[j] ? x : 0;
        }
    }

    // y_mask: write once (only channel-group 0), vectorized
    if (cgroup == 0) {
        const int ylen = ylen_ws[b];
        float4 m;
        m.x = (y0 + 0 < ylen) ? 1.0f : 0.0f;
        m.y = (y0 + 1 < ylen) ? 1.0f : 0.0f;
        m.z = (y0 + 2 < ylen) ? 1.0f : 0.0f;
        m.w = (y0 + 3 < ylen) ? 1.0f : 0.0f;
        *(float4*)(mask_out + (size_t)b * TY_ + y0) = m;
    }

    // --- expand stats and apply noise -----------------------------------
    const int c0 = cgroup * CPER;
    for (int c = 0; c < CPER; ++c) {
        const size_t row = (size_t)(b * C_ + c0 + c);
        const size_t sb  = row * TX_;
        const size_t ni  = row * TY_ + (size_t)y0;

        if (c + 1 < CPER) {             // global_prefetch_b8 next channel's stats
            __builtin_prefetch(&m_p[sb + TX_ + xs[0]],    0, 1);
            __builtin_prefetch(&logs_p[sb + TX_ + xs[0]], 0, 1);
        }

        const float4 nz = *(const float4*)(noise + ni);
        float4 z;
        z.x = vs[0] ? m_p[sb + xs[0]] + nz.x * expf(logs_p[sb + xs[0]]) : nz.x;
        z.y = vs[1] ? m_p[sb + xs[1]] + nz.y * expf(logs_p[sb + xs[1]]) : nz.y;
        z.z = vs[2] ? m_p[sb + xs[2]] + nz.z * expf(logs_p[sb + xs[2]]) : nz.z;
        z.w = vs[3] ? m_p[sb + xs[3]] + nz.w * expf(logs_p[sb + xs[3]]) : nz.w;
        *(float4*)(z_out + ni) = z;
    }
}

// ---------------------------------------------------------------------------
extern "C" void kernel_launch(void* const* d_in, const int* in_sizes, int n_in,
                              void* d_out, int out_size, void* d_ws, size_t ws_size,
                              hipStream_t stream) {
    const float* m_p       = (const float*)d_in[0];  // [B,C,TX]
    const float* logs_p    = (const float*)d_in[1];  // [B,C,TX]
    const float* logw      = (const float*)d_in[2];  // [B,1,TX]
    const float* noise     = (const float*)d_in[3];  // [B,C,TY]
    const int*   x_lengths = (const int*)d_in[4];    // [B]
    const int*   max_y     = (const int*)d_in[5];    // [1]

    float* z_out    = (float*)d_out;                             // [B,C,TY]
    float* mask_out = (float*)d_out + (size_t)B_ * C_ * TY_;     // [B,TY]

    float* cum_ws  = (float*)d_ws;                               // [B,TX]
    int*   ylen_ws = (int*)((char*)d_ws + (size_t)B_ * TX_ * sizeof(float));

    vits_scan_kernel<<<B_, TX_, 0, stream>>>(logw, x_lengths, max_y, cum_ws, ylen_ws);

    dim3 grid(B_ * TILES_PER_B, CSPLIT, 1);
    vits_expand_kernel<<<grid, TBLK, 0, stream>>>(m_p, logs_p, noise, cum_ws,
                                                  ylen_ws, z_out, mask_out);
}